// SimilaritySelection_20323785245157
// MI455X (gfx1250) — compile-verified
//
#include <hip/hip_runtime.h>

// Problem dims (fixed by the reference)
#define S_DIM 16
#define B_DIM 512
#define H_DIM 2048
#define EPS_F 1e-8f

typedef __attribute__((ext_vector_type(2))) float v2f;
typedef __attribute__((ext_vector_type(8))) float v8f;

// ---------------------------------------------------------------------------
// Kernel 1: o_norm2[b] = sum_h OG[b,h]^2           (512 blocks x 256 threads)
// ---------------------------------------------------------------------------
__global__ void onorm2_kernel(const float* __restrict__ og,
                              float* __restrict__ onorm2) {
  __shared__ float red[256];
  const int b = blockIdx.x;
  const float* row = og + (size_t)b * H_DIM;
  float acc = 0.0f;
  for (int h = threadIdx.x; h < H_DIM; h += 256) {
    float x = row[h];
    acc += x * x;
  }
  red[threadIdx.x] = acc;
  __syncthreads();
  for (int off = 128; off > 0; off >>= 1) {
    if (threadIdx.x < off) red[threadIdx.x] += red[threadIdx.x + off];
    __syncthreads();
  }
  if (threadIdx.x == 0) onorm2[b] = red[0];
}

// ---------------------------------------------------------------------------
// Kernel 2: sims[s*B+b] = dot(enh[s,b], OG[b]) / max(|enh|*|OG|, eps)
//           (8192 blocks x 256 threads, fused dot + enh-norm reduction)
// ---------------------------------------------------------------------------
__global__ void sims_kernel(const float* __restrict__ og,
                            const float* __restrict__ enh,
                            const float* __restrict__ onorm2,
                            float* __restrict__ sims) {
  __shared__ float rdot[256];
  __shared__ float ren[256];
  const int sb = blockIdx.x;       // s*B + b
  const int b  = sb % B_DIM;
  const float* erow = enh + (size_t)sb * H_DIM;
  const float* orow = og  + (size_t)b  * H_DIM;
  float dot = 0.0f, en2 = 0.0f;
  for (int h = threadIdx.x; h < H_DIM; h += 256) {
    float e = erow[h];
    float o = orow[h];
    dot += e * o;
    en2 += e * e;
  }
  rdot[threadIdx.x] = dot;
  ren[threadIdx.x]  = en2;
  __syncthreads();
  for (int off = 128; off > 0; off >>= 1) {
    if (threadIdx.x < off) {
      rdot[threadIdx.x] += rdot[threadIdx.x + off];
      ren[threadIdx.x]  += ren[threadIdx.x + off];
    }
    __syncthreads();
  }
  if (threadIdx.x == 0) {
    float denom = fmaxf(sqrtf(ren[0]) * sqrtf(onorm2[b]), EPS_F);
    sims[sb] = rdot[0] / denom;
  }
}

// ---------------------------------------------------------------------------
// Kernel 3: wrep[b,h] = (1/S) * sum_s sims[s,b] * enh[s,b,h]
//           (elementwise over B*H, coalesced in h)
// ---------------------------------------------------------------------------
__global__ void wrep_kernel(const float* __restrict__ enh,
                            const float* __restrict__ sims,
                            float* __restrict__ wrep) {
  const int idx = blockIdx.x * 256 + threadIdx.x;   // b*H + h
  const int b = idx / H_DIM;
  float acc = 0.0f;
#pragma unroll
  for (int s = 0; s < S_DIM; ++s) {
    acc += sims[s * B_DIM + b] * enh[(size_t)s * (B_DIM * H_DIM) + idx];
  }
  wrep[idx] = acc * (1.0f / (float)S_DIM);
}

// ---------------------------------------------------------------------------
// Kernel 4: out[b,j] = sum_h wrep[b,h] * W[j,h] + bias[j]
// f32 WMMA GEMM (16x16x4). One wave -> 16(M) x 64(N) tile via 4 accumulators
// (A loaded once per K-step, reused for 4 B-tiles).
//   A 16x4 f32 layout: lane = {r = lane&15 -> row M, kh = lane>>4 -> K pair},
//     v0 = A[r][2*kh], v1 = A[r][2*kh+1]   -> one 8-byte v2f load.
//   B 4x16 f32 layout mirrors A with rows of W^T = columns h of W[j,:].
// Tile grid: 32 M-tiles x 32 N-groups = 1024 waves = 128 blocks x 8 waves,
// exact cover -> EXEC all-1s at every WMMA.
// ---------------------------------------------------------------------------
__global__ void gemm_wmma_kernel(const float* __restrict__ wrep,
                                 const float* __restrict__ W,
                                 const float* __restrict__ bias,
                                 float* __restrict__ out) {
  const int lane = threadIdx.x & 31;
  const int wave = threadIdx.x >> 5;
  const int task = blockIdx.x * 8 + wave;  // 0..1023
  const int mt = task >> 5;                // 0..31  -> b tile
  const int ng = task & 31;                // 0..31  -> group of 4 n tiles
  const int b0 = mt * 16;
  const int j0 = ng * 64;
  const int r  = lane & 15;                // row within tile
  const int kh = lane >> 4;                // selects K pair {2kh, 2kh+1}

  const float* aptr = wrep + (size_t)(b0 + r) * H_DIM + 2 * kh;
  const float* bptr = W    + (size_t)(j0 + r) * H_DIM + 2 * kh;

  v8f acc0 = {}, acc1 = {}, acc2 = {}, acc3 = {};

#pragma unroll 4
  for (int k0 = 0; k0 < H_DIM; k0 += 4) {
    v2f a  = *(const v2f*)(aptr + k0);
    v2f w0 = *(const v2f*)(bptr + k0);
    v2f w1 = *(const v2f*)(bptr + (size_t)16 * H_DIM + k0);
    v2f w2 = *(const v2f*)(bptr + (size_t)32 * H_DIM + k0);
    v2f w3 = *(const v2f*)(bptr + (size_t)48 * H_DIM + k0);
    acc0 = __builtin_amdgcn_wmma_f32_16x16x4_f32(false, a, false, w0, (short)0, acc0, false, false);
    acc1 = __builtin_amdgcn_wmma_f32_16x16x4_f32(false, a, false, w1, (short)0, acc1, false, false);
    acc2 = __builtin_amdgcn_wmma_f32_16x16x4_f32(false, a, false, w2, (short)0, acc2, false, false);
    acc3 = __builtin_amdgcn_wmma_f32_16x16x4_f32(false, a, false, w3, (short)0, acc3, false, false);
  }

  // C/D layout: VGPR v, lanes 0-15 -> (M=v, N=r); lanes 16-31 -> (M=v+8, N=r)
  const float bias0 = bias[j0 + r];
  const float bias1 = bias[j0 + 16 + r];
  const float bias2 = bias[j0 + 32 + r];
  const float bias3 = bias[j0 + 48 + r];
  const int row_base = b0 + 8 * kh;
#pragma unroll
  for (int v = 0; v < 8; ++v) {
    float* orow = out + (size_t)(row_base + v) * H_DIM + j0 + r;
    orow[0]  = acc0[v] + bias0;
    orow[16] = acc1[v] + bias1;
    orow[32] = acc2[v] + bias2;
    orow[48] = acc3[v] + bias3;
  }
}

// ---------------------------------------------------------------------------
// Launch
// ---------------------------------------------------------------------------
extern "C" void kernel_launch(void* const* d_in, const int* in_sizes, int n_in,
                              void* d_out, int out_size, void* d_ws, size_t ws_size,
                              hipStream_t stream) {
  const float* og   = (const float*)d_in[0];  // (B, H)
  const float* enh  = (const float*)d_in[1];  // (S, B, H)
  const float* W    = (const float*)d_in[2];  // (H, H)
  const float* bias = (const float*)d_in[3];  // (H,)
  float* out = (float*)d_out;                 // (B, H)

  float* ws     = (float*)d_ws;
  float* onorm2 = ws;                  // 512 floats
  float* sims   = ws + 512;            // 8192 floats (S*B)
  float* wrep   = ws + 512 + 8192;     // 512*2048 floats; byte off 34816 (16-aligned)

  onorm2_kernel<<<B_DIM, 256, 0, stream>>>(og, onorm2);
  sims_kernel<<<S_DIM * B_DIM, 256, 0, stream>>>(og, enh, onorm2, sims);
  wrep_kernel<<<(B_DIM * H_DIM) / 256, 256, 0, stream>>>(enh, sims, wrep);
  gemm_wmma_kernel<<<128, 256, 0, stream>>>(wrep, W, bias, out);
}